// MultivariateGaussianMDN_16441134809788
// MI455X (gfx1250) — compile-verified
//
#include <hip/hip_runtime.h>
#include <cstddef>

// ---- problem constants (from reference) ----
#define F_DIM 10
#define KMIX  20
#define HDIM  512
#define CDIM  128
#define UPN   45            // F*(F-1)/2
#define BROWS 32            // batch rows per block = two 16-row WMMA M-tiles
#define NCAT  1320          // 20 + 200 + 200 + 900 concatenated output columns
#define LOG2PI 1.837877066409345f

// LDS padded strides: pad of 4 floats -> stride % 64 == 4 (44 for OPAD) banks per row step,
// conflict-free for the access patterns below, and keeps 8/16-byte alignment.
#define CPAD (CDIM + 4)     // 132
#define HPAD (HDIM + 4)     // 516
#define OPAD (NCAT + 4)     // 1324

typedef __attribute__((ext_vector_type(2))) float v2f;
typedef __attribute__((ext_vector_type(8))) float v8f;

__device__ __forceinline__ v8f wmma_f32(v2f a, v2f b, v8f c) {
    return __builtin_amdgcn_wmma_f32_16x16x4_f32(false, a, false, b,
                                                 (short)0, c, false, false);
}

// Native CDNA5 transcendental (V_TANH_F32, gfx1250). Two v_nop after a TRANS
// op per the ISA multicycle co-execution hazard rule, since the compiler
// cannot schedule around the asm block.
__device__ __forceinline__ float tanh_hw(float x) {
    float r;
    asm volatile("v_tanh_f32 %0, %1\n\tv_nop\n\tv_nop" : "=v"(r) : "v"(x));
    return r;
}

__global__ __launch_bounds__(256)
void mdn_fused_kernel(const float* __restrict__ inputs,
                      const float* __restrict__ context,
                      const float* __restrict__ Wh, const float* __restrict__ bh,
                      const float* __restrict__ Wl, const float* __restrict__ bl,
                      const float* __restrict__ Wm, const float* __restrict__ bm,
                      const float* __restrict__ Wd, const float* __restrict__ bd,
                      const float* __restrict__ Wu, const float* __restrict__ bu,
                      float* __restrict__ out, int B)
{
    __shared__ float s_ctx[BROWS][CPAD];   //  16.5 KB : context tile
    __shared__ float s_h  [BROWS][HPAD];   //  64.5 KB : tanh hidden tile
    __shared__ float s_out[BROWS][OPAD];   // 165.5 KB : [logits|means|diag|upper]
    __shared__ float s_term[BROWS][KMIX + 4];

    const int tid  = threadIdx.x;
    const int lane = tid & 31;
    const int wave = tid >> 5;
    const int half = lane >> 4;       // 0: lanes 0-15, 1: lanes 16-31
    const int l16  = lane & 15;
    const int row0 = blockIdx.x * BROWS;

    // ---------- phase 0: stage context tile (vectorized) ----------
    for (int i = tid; i < BROWS * (CDIM / 4); i += 256) {
        const int r = i / (CDIM / 4), c4 = i % (CDIM / 4);
        int rg = row0 + r; if (rg >= B) rg = B - 1;     // clamp tail block
        const float4 v = reinterpret_cast<const float4*>(
                             context + (size_t)rg * CDIM)[c4];
        *reinterpret_cast<float4*>(&s_ctx[r][c4 * 4]) = v;
    }
    __syncthreads();

    // ---------- phase 1: h = tanh(ctx @ Wh + bh) ----------
    // 32 N-tiles over 8 waves; each wave carries TWO 16-row accumulators so
    // every Wh fragment load feeds two WMMAs.
    for (int t = wave; t < HDIM / 16; t += 8) {
        const int n0 = t * 16;
        v8f acc0 = {}, acc1 = {};
        #pragma unroll 4
        for (int k0 = 0; k0 < CDIM; k0 += 4) {
            const int ka = k0 + 2 * half;          // K index of component .x
            v2f bf, a0, a1;
            bf.x = Wh[(size_t)ka * HDIM + n0 + l16];
            bf.y = Wh[(size_t)(ka + 1) * HDIM + n0 + l16];
            a0.x = s_ctx[l16     ][ka];  a0.y = s_ctx[l16     ][ka + 1];
            a1.x = s_ctx[l16 + 16][ka];  a1.y = s_ctx[l16 + 16][ka + 1];
            acc0 = wmma_f32(a0, bf, acc0);
            acc1 = wmma_f32(a1, bf, acc1);
        }
        const float bias = bh[n0 + l16];
        #pragma unroll
        for (int j = 0; j < 8; ++j) {
            const int m = j + 8 * half;            // C/D row mapping
            s_h[m     ][n0 + l16] = tanh_hw(acc0[j] + bias);
            s_h[m + 16][n0 + l16] = tanh_hw(acc1[j] + bias);
        }
    }
    __syncthreads();

    // ---------- phase 2: [logits|means|diag|upper] = h @ Wcat + bcat ----------
    // 85 column tiles: Wl(20)->2, Wm(200)->13, Wd(200)->13, Wu(900)->57
    for (int t = wave; t < 85; t += 8) {
        const float *W, *bias;
        int ncols, cbase, lt;
        if      (t < 2)  { W = Wl; bias = bl; ncols = 20;  cbase = 0;   lt = t;      }
        else if (t < 15) { W = Wm; bias = bm; ncols = 200; cbase = 20;  lt = t - 2;  }
        else if (t < 28) { W = Wd; bias = bd; ncols = 200; cbase = 220; lt = t - 15; }
        else             { W = Wu; bias = bu; ncols = 900; cbase = 420; lt = t - 28; }
        const int  n0    = lt * 16;
        const int  n     = n0 + l16;
        const bool valid = (n < ncols);            // VALU mask; EXEC stays full for WMMA
        v8f acc0 = {}, acc1 = {};
        #pragma unroll 8
        for (int k0 = 0; k0 < HDIM; k0 += 4) {
            const int ka = k0 + 2 * half;
            v2f bf, a0, a1;
            bf.x = valid ? W[(size_t)ka * ncols + n]       : 0.0f;
            bf.y = valid ? W[(size_t)(ka + 1) * ncols + n] : 0.0f;
            a0.x = s_h[l16     ][ka];  a0.y = s_h[l16     ][ka + 1];
            a1.x = s_h[l16 + 16][ka];  a1.y = s_h[l16 + 16][ka + 1];
            acc0 = wmma_f32(a0, bf, acc0);
            acc1 = wmma_f32(a1, bf, acc1);
        }
        if (valid) {
            const float bv = bias[n];
            #pragma unroll
            for (int j = 0; j < 8; ++j) {
                const int m = j + 8 * half;
                s_out[m     ][cbase + n] = acc0[j] + bv;
                s_out[m + 16][cbase + n] = acc1[j] + bv;
            }
        }
    }
    __syncthreads();

    // ---------- phase 3: per-(row,k) MoG term ----------
    // d1^T (A^T A) d1 = ||A d1||^2 with A upper-triangular: 55 FMAs per (r,k).
    for (int p = tid; p < BROWS * KMIX; p += 256) {
        const int r = p / KMIX;
        const int k = p % KMIX;
        int rg = row0 + r; if (rg >= B) rg = B - 1;
        const float* mrow = &s_out[r][20  + k * F_DIM];   // means
        const float* drow = &s_out[r][220 + k * F_DIM];   // pre-softplus diag
        const float* urow = &s_out[r][420 + k * UPN];     // strict upper entries
        const float* inrow = inputs + (size_t)rg * F_DIM;
        float d1[F_DIM];
        #pragma unroll
        for (int f = 0; f < F_DIM; ++f) d1[f] = inrow[f] - mrow[f];
        float sld = 0.0f, quad = 0.0f;
        int uoff = 0;
        #pragma unroll
        for (int j = 0; j < F_DIM; ++j) {
            const float x  = drow[j];
            const float dj = fmaxf(x, 0.0f) + log1pf(expf(-fabsf(x))) + 0.01f; // softplus+EPS
            float y = dj * d1[j];                                // y = (A d1)_j
            for (int c = j + 1; c < F_DIM; ++c)
                y = fmaf(urow[uoff + (c - j - 1)], d1[c], y);
            uoff += (F_DIM - 1 - j);
            quad = fmaf(y, y, quad);
            sld += logf(dj);
        }
        s_term[r][k] = s_out[r][k] + sld - 0.5f * quad;
    }
    __syncthreads();

    // ---------- phase 4: dual logsumexp per row ----------
    // logsumexp(log_softmax(l)+b+sld+d) = b + LSE(l+sld+d) - LSE(l)
    if (tid < BROWS) {
        const int r  = tid;
        const int rg = row0 + r;
        if (rg < B) {
            float mt = -INFINITY, ml = -INFINITY;
            #pragma unroll
            for (int k = 0; k < KMIX; ++k) {
                mt = fmaxf(mt, s_term[r][k]);
                ml = fmaxf(ml, s_out[r][k]);      // raw logits
            }
            float st = 0.0f, sl = 0.0f;
            #pragma unroll
            for (int k = 0; k < KMIX; ++k) {
                st += expf(s_term[r][k] - mt);
                sl += expf(s_out[r][k] - ml);
            }
            out[rg] = (mt + logf(st)) - (ml + logf(sl))
                      - 0.5f * (float)F_DIM * LOG2PI;
        }
    }
}

extern "C" void kernel_launch(void* const* d_in, const int* in_sizes, int n_in,
                              void* d_out, int out_size, void* d_ws, size_t ws_size,
                              hipStream_t stream) {
    const float* inputs  = (const float*)d_in[0];
    const float* context = (const float*)d_in[1];
    const float* Wh = (const float*)d_in[2];
    const float* bh = (const float*)d_in[3];
    const float* Wl = (const float*)d_in[4];
    const float* bl = (const float*)d_in[5];
    const float* Wm = (const float*)d_in[6];
    const float* bm = (const float*)d_in[7];
    const float* Wd = (const float*)d_in[8];
    const float* bd = (const float*)d_in[9];
    const float* Wu = (const float*)d_in[10];
    const float* bu = (const float*)d_in[11];

    const int B    = in_sizes[1] / CDIM;      // context is [B, 128]
    const int grid = (B + BROWS - 1) / BROWS; // 1563 blocks for B=50000

    mdn_fused_kernel<<<grid, 256, 0, stream>>>(inputs, context, Wh, bh, Wl, bl,
                                               Wm, bm, Wd, bd, Wu, bu,
                                               (float*)d_out, B);
}